// MultiHeadedAttention_77163382440919
// MI455X (gfx1250) — compile-verified
//
#include <hip/hip_runtime.h>
#include <hip/hip_bf16.h>

// MI455X / gfx1250, wave32. fp32 WMMA (16x16x4) everywhere: problem is
// bandwidth-bound (~670MB traffic vs 7.5 GFLOP) and prob_attn is an fp32
// output, so low-precision matrix ops buy nothing and break fidelity.
// Proj GEMM stages tiles into LDS with the Tensor Data Mover (TDM) using the
// descriptor pad feature to get a bank-conflict-free stride-33 layout.

typedef __attribute__((ext_vector_type(2))) float v2f;
typedef __attribute__((ext_vector_type(8))) float v8f;
typedef __attribute__((ext_vector_type(4))) unsigned int u32x4;
typedef __attribute__((ext_vector_type(4))) int i32x4;
typedef __attribute__((ext_vector_type(8))) int i32x8;

#define BATCH 16
#define SEQ   512
#define DIM   256
#define HEADS 8
#define DH    32

static __device__ __forceinline__ v8f wmma_f32(v2f a, v2f b, v8f c) {
    return __builtin_amdgcn_wmma_f32_16x16x4_f32(
        /*neg_a=*/false, a, /*neg_b=*/false, b,
        /*c_mod=*/(short)0, c, /*reuse_a=*/false, /*reuse_b=*/false);
}

#if __has_builtin(__builtin_amdgcn_tensor_load_to_lds)
#define USE_TDM 1
#else
#define USE_TDM 0
#endif

#if USE_TDM
// 2D TDM load: tile (tile0 x tile1) of 4-byte elements, row stride `stride0`
// elements, DMA'd from global to LDS with +1 DWORD pad every 32 DWORDs
// (=> LDS row stride 33 floats, bank-conflict free for fragment reads).
// Descriptor layout per CDNA5 ISA 8.3/8.4.
static __device__ __forceinline__ void tdm_load_2d(unsigned int lds_off,
                                                   const void* gptr,
                                                   unsigned int tile0,
                                                   unsigned int tile1,
                                                   unsigned long long stride0,
                                                   unsigned int td0,
                                                   unsigned int td1) {
    unsigned long long ga = (unsigned long long)(uintptr_t)gptr;
    u32x4 g0;
    g0[0] = 1u;                                        // count=1 (valid), user desc
    g0[1] = lds_off;                                   // lds_addr [63:32]
    g0[2] = (unsigned int)ga;                          // global_addr low
    g0[3] = (unsigned int)((ga >> 32) & 0x1FFFFFFu)    // global_addr[56:32]
          | (2u << 30);                                // type = 2 ("image")
    i32x8 g1;
    g1[0] = (int)((2u << 16)      // data_size = 4 bytes
                | (1u << 20)      // pad_enable
                | (4u << 22));    // pad_interval: every 32 DWORDs (+pad_amount=1 DWORD)
    g1[1] = (int)((td0 & 0xFFFFu) << 16);              // tensor_dim0[15:0]
    g1[2] = (int)((td0 >> 16) | ((td1 & 0xFFFFu) << 16)); // td0 hi | td1 lo
    g1[3] = (int)((td1 >> 16) | (tile0 << 16));        // td1 hi | tile_dim0
    g1[4] = (int)(tile1 & 0xFFFFu);                    // tile_dim1 | tile_dim2=0
    g1[5] = (int)(unsigned int)stride0;                // tensor_dim0_stride[31:0]
    g1[6] = (int)((stride0 >> 32) & 0xFFFFu);          // stride0 hi | stride1 lo = 0
    g1[7] = 0;
    i32x4 z4 = {0, 0, 0, 0};
#if __clang_major__ >= 23
    i32x8 z8 = {0, 0, 0, 0, 0, 0, 0, 0};
    __builtin_amdgcn_tensor_load_to_lds(g0, g1, z4, z4, z8, 0);
#else
    __builtin_amdgcn_tensor_load_to_lds(g0, g1, z4, z4, 0);
#endif
}

static __device__ __forceinline__ void tdm_wait() {
#if __has_builtin(__builtin_amdgcn_s_wait_tensorcnt)
    __builtin_amdgcn_s_wait_tensorcnt(0);
#else
    asm volatile("s_wait_tensorcnt 0x0" ::: "memory");
#endif
}
#endif  // USE_TDM

// ---------------------------------------------------------------------------
// Kernel 1: fused Q/K/V projection.  out[n,j] = sum_k X[n,k]*W[j,k] + b[j],
// scattered into (b,h,s,dh).  grid=(M/16, N/128, 3), block=256 (8 waves).
// A tile (16x32) and W tile (128x32) staged in LDS (stride 33) via TDM.
// B-fragment is read from the row-major W tile: B[k][col] = lds_w[col*33+k].
// ---------------------------------------------------------------------------
__global__ void proj_kernel(const float* __restrict__ query,
                            const float* __restrict__ key,
                            const float* __restrict__ value,
                            const float* __restrict__ Wq,
                            const float* __restrict__ Wk,
                            const float* __restrict__ Wv,
                            const float* __restrict__ bq,
                            const float* __restrict__ bk,
                            const float* __restrict__ bv,
                            float* __restrict__ qws,
                            float* __restrict__ kws,
                            float* __restrict__ vws) {
    const int z = blockIdx.z;
    const float* X    = (z == 0) ? query : (z == 1) ? key : value;
    const float* W    = (z == 0) ? Wq    : (z == 1) ? Wk  : Wv;
    const float* bias = (z == 0) ? bq    : (z == 1) ? bk  : bv;
    float*       O    = (z == 0) ? qws   : (z == 1) ? kws : vws;

    const int rowBase = blockIdx.x * 16;    // M tile (n = b*S+s)
    const int jBase   = blockIdx.y * 128;   // N super-tile (output feature)

    const int tid  = threadIdx.x;
    const int wave = tid >> 5;
    const int lane = tid & 31;
    const int l15  = lane & 15;
    const int hi   = (lane >> 4) & 1;       // 0: K pair {0,1}, 1: K pair {2,3}
    const int koff = hi * 2;

    __shared__ float lds_a[16 * 33];        // 16 rows x 32 K, stride 33
    __shared__ float lds_w[128 * 33];       // 128 j-rows x 32 K, stride 33

    v8f acc = {0.f, 0.f, 0.f, 0.f, 0.f, 0.f, 0.f, 0.f};

#if USE_TDM
    const unsigned int lds_a_off = (unsigned int)(uintptr_t)(void*)lds_a;
    const unsigned int lds_w_off = (unsigned int)(uintptr_t)(void*)lds_w;
#endif

    for (int k0 = 0; k0 < DIM; k0 += 32) {
        __syncthreads();   // previous chunk fully consumed before overwrite
#if USE_TDM
        if (wave == 0) {
            tdm_load_2d(lds_a_off, X + (size_t)rowBase * DIM + k0,
                        /*tile*/ 32, 16, /*stride*/ DIM, /*tensor*/ 32, 16);
            tdm_load_2d(lds_w_off, W + (size_t)jBase * DIM + k0,
                        /*tile*/ 32, 128, /*stride*/ DIM, /*tensor*/ 32, 128);
            tdm_wait();
        }
#else
        for (int i = tid; i < 16 * 32; i += 256) {
            int m = i >> 5, kk = i & 31;
            lds_a[m * 33 + kk] = X[(size_t)(rowBase + m) * DIM + k0 + kk];
        }
        for (int i = tid; i < 128 * 32; i += 256) {
            int j = i >> 5, kk = i & 31;
            lds_w[j * 33 + kk] = W[(size_t)(jBase + j) * DIM + k0 + kk];
        }
#endif
        __syncthreads();

        if ((k0 + 32) < DIM)
            __builtin_prefetch(X + (size_t)(rowBase + l15) * DIM + k0 + 32, 0, 0);

#pragma unroll
        for (int kk = 0; kk < 32; kk += 4) {
            v2f a, b;
            a.x = lds_a[l15 * 33 + kk + koff];
            a.y = lds_a[l15 * 33 + kk + koff + 1];
            b.x = lds_w[(wave * 16 + l15) * 33 + kk + koff];
            b.y = lds_w[(wave * 16 + l15) * 33 + kk + koff + 1];
            acc = wmma_f32(a, b, acc);
        }
    }

    // scatter D tile into (b,h,s,dh) layout, adding bias
    const int j  = jBase + wave * 16 + l15;
    const int h  = j >> 5;
    const int dh = j & 31;
    const float bj = bias[j];
#pragma unroll
    for (int r = 0; r < 8; ++r) {
        int m = rowBase + r + hi * 8;      // global row n
        int bb = m >> 9;
        int s  = m & 511;
        O[(((size_t)bb * HEADS + h) * SEQ + s) * DH + dh] = acc[r] + bj;
    }
}

// ---------------------------------------------------------------------------
// block reduction helpers (256 threads)
// ---------------------------------------------------------------------------
static __device__ __forceinline__ float block_max(float v, float* sh) {
    int tid = threadIdx.x;
    sh[tid] = v;
    __syncthreads();
    for (int s = 128; s > 0; s >>= 1) {
        if (tid < s) sh[tid] = fmaxf(sh[tid], sh[tid + s]);
        __syncthreads();
    }
    float r = sh[0];
    __syncthreads();
    return r;
}
static __device__ __forceinline__ float block_sum(float v, float* sh) {
    int tid = threadIdx.x;
    sh[tid] = v;
    __syncthreads();
    for (int s = 128; s > 0; s >>= 1) {
        if (tid < s) sh[tid] += sh[tid + s];
        __syncthreads();
    }
    float r = sh[0];
    __syncthreads();
    return r;
}

// ---------------------------------------------------------------------------
// Kernel 2: head-independent blend term
//   base[b,q,k] = (1-l1)*l2*time_attn + l1*rel_attn
// grid = B*S rows, block = 256, 2 cols/thread. Max-subtracted softmaxes to
// match JAX (rows that are entirely -1e4 must become uniform).
// ---------------------------------------------------------------------------
__global__ void base_kernel(const float* __restrict__ rel,
                            const float* __restrict__ ts,
                            const float* __restrict__ l1p,
                            const float* __restrict__ l2p,
                            float* __restrict__ base) {
    const int rowid = blockIdx.x;            // b*S + q
    const int q = rowid & (SEQ - 1);
    const int tid = threadIdx.x;
    const float l1 = *l1p, l2 = *l2p;

    const float* relrow = rel + (size_t)rowid * SEQ;
    const float* tsrow  = ts  + (size_t)rowid * SEQ;

    float relLogit[2], tsLogit[2];
#pragma unroll
    for (int i = 0; i < 2; ++i) {
        int k = tid + i * 256;
        bool m = k > q;                       // mask == "future" position
        float rf = m ? relrow[k] : 0.0f;      // relf = rel * mask
        relLogit[i] = (rf == 0.0f) ? -10000.0f : rf;
        float t = expf(-fabsf(tsrow[k]));
        tsLogit[i] = m ? -__builtin_inff() : t;
    }

    __shared__ float sh[256];
    float relMax = block_max(fmaxf(relLogit[0], relLogit[1]), sh);
    float tsMax  = block_max(fmaxf(tsLogit[0], tsLogit[1]), sh);

    float relE[2], tsE[2];
#pragma unroll
    for (int i = 0; i < 2; ++i) {
        relE[i] = expf(relLogit[i] - relMax);
        tsE[i]  = expf(tsLogit[i] - tsMax);   // exp(-inf - m) == 0
    }
    float relSum = block_sum(relE[0] + relE[1], sh);
    float tsSum  = block_sum(tsE[0] + tsE[1], sh);

    const float wT = (1.0f - l1) * l2;
#pragma unroll
    for (int i = 0; i < 2; ++i) {
        int k = tid + i * 256;
        base[(size_t)rowid * SEQ + k] = wT * (tsE[i] / tsSum) + l1 * (relE[i] / relSum);
    }
}

// ---------------------------------------------------------------------------
// Kernel 3: scores = q.k^T / sqrt(DH), causal mask, written to prob region.
// grid = (4 k-supertiles, 32 q-tiles, B*H), block = 256 (8 waves, each a
// 16x16 tile, K-loop over DH=32 -> 8 WMMAs).
// ---------------------------------------------------------------------------
__global__ void score_kernel(const float* __restrict__ qws,
                             const float* __restrict__ kws,
                             float* __restrict__ prob) {
    const int bh    = blockIdx.z;
    const int qBase = blockIdx.y * 16;
    const int tid   = threadIdx.x;
    const int wave  = tid >> 5;
    const int lane  = tid & 31;
    const int l15   = lane & 15;
    const int hi    = (lane >> 4) & 1;
    const int koff  = hi * 2;
    const int kBase = blockIdx.x * 128 + wave * 16;

    const int row = qBase + l15;
    const int col = kBase + l15;
    const float* ap0 = qws + ((size_t)bh * SEQ + row) * DH;
    const float* bp0 = kws + ((size_t)bh * SEQ + col) * DH;

    v8f acc = {0.f, 0.f, 0.f, 0.f, 0.f, 0.f, 0.f, 0.f};
#pragma unroll
    for (int kk = 0; kk < DH; kk += 4) {
        v2f a, b;
        a.x = ap0[kk + koff];
        a.y = ap0[kk + koff + 1];
        b.x = bp0[kk + koff];
        b.y = bp0[kk + koff + 1];
        acc = wmma_f32(a, b, acc);
    }

    const float scale = 0.17677669529663687f;   // 1/sqrt(32)
#pragma unroll
    for (int r = 0; r < 8; ++r) {
        int m = qBase + r + hi * 8;
        int c = kBase + l15;
        float v = acc[r] * scale;
        if (c > m) v = -1.0e9f;                 // causal mask
        prob[((size_t)bh * SEQ + m) * SEQ + c] = v;
    }
}

// ---------------------------------------------------------------------------
// Kernel 4: in-place row softmax of scores + blend with base.
// grid = B*H*S, block = 256, 2 cols/thread.
// ---------------------------------------------------------------------------
__global__ void softmax_combine_kernel(float* __restrict__ prob,
                                       const float* __restrict__ base,
                                       const float* __restrict__ l1p,
                                       const float* __restrict__ l2p) {
    const int rowid = blockIdx.x;          // bh*S + q
    const int bh = rowid >> 9;
    const int q  = rowid & (SEQ - 1);
    const int b  = bh >> 3;
    const int tid = threadIdx.x;
    const float l1 = *l1p, l2 = *l2p;
    const float c = (1.0f - l1) * (1.0f - l2);

    float* prow = prob + (size_t)rowid * SEQ;
    const float* brow = base + ((size_t)b * SEQ + q) * SEQ;

    float v[2];
    v[0] = prow[tid];
    v[1] = prow[tid + 256];

    __shared__ float sh[256];
    float mx = block_max(fmaxf(v[0], v[1]), sh);
    float e[2];
    e[0] = expf(v[0] - mx);
    e[1] = expf(v[1] - mx);
    float sum = block_sum(e[0] + e[1], sh);
    float inv = 1.0f / sum;

    prow[tid]       = c * e[0] * inv + brow[tid];
    prow[tid + 256] = c * e[1] * inv + brow[tid + 256];
}

// ---------------------------------------------------------------------------
// Kernel 5: out = prob @ v, un-transposed back to (b,s,d).
// grid = (4 q-supertiles, 2 n-tiles, B*H), block = 256 (8 waves).
// Each wave: 16x16 tile, K-loop 512 -> 128 WMMAs.
// ---------------------------------------------------------------------------
__global__ void out_kernel(const float* __restrict__ prob,
                           const float* __restrict__ vws,
                           float* __restrict__ out) {
    const int bh = blockIdx.z;
    const int b  = bh >> 3;
    const int h  = bh & 7;
    const int nBase = blockIdx.y * 16;
    const int tid  = threadIdx.x;
    const int wave = tid >> 5;
    const int lane = tid & 31;
    const int l15  = lane & 15;
    const int hi   = (lane >> 4) & 1;
    const int koff = hi * 2;
    const int qBase = (blockIdx.x * 8 + wave) * 16;

    const int row  = qBase + l15;
    const int ncol = nBase + l15;
    const float* prow = prob + ((size_t)bh * SEQ + row) * SEQ;
    const float* vbase = vws + (size_t)bh * SEQ * DH + ncol;

    v8f acc = {0.f, 0.f, 0.f, 0.f, 0.f, 0.f, 0.f, 0.f};
#pragma unroll 8
    for (int kk = 0; kk < SEQ; kk += 4) {
        if ((kk & 63) == 0 && kk + 128 < SEQ)
            __builtin_prefetch(prow + kk + 128, 0, 0);
        v2f a, bf;
        a.x = prow[kk + koff];
        a.y = prow[kk + koff + 1];
        const float* bp = vbase + (size_t)(kk + koff) * DH;
        bf.x = bp[0];
        bf.y = bp[DH];
        acc = wmma_f32(a, bf, acc);
    }

#pragma unroll
    for (int r = 0; r < 8; ++r) {
        int m = qBase + r + hi * 8;        // sequence position
        out[((size_t)b * SEQ + m) * DIM + h * DH + nBase + l15] = acc[r];
    }
}

// ---------------------------------------------------------------------------
extern "C" void kernel_launch(void* const* d_in, const int* in_sizes, int n_in,
                              void* d_out, int out_size, void* d_ws, size_t ws_size,
                              hipStream_t stream) {
    const float* query = (const float*)d_in[0];
    const float* key_t = (const float*)d_in[1];
    const float* value = (const float*)d_in[2];
    const float* rel   = (const float*)d_in[3];
    const float* tstmp = (const float*)d_in[4];
    // d_in[5] = mask: deterministically (k > q), recomputed in-kernel
    const float* l1    = (const float*)d_in[6];
    const float* l2    = (const float*)d_in[7];
    const float* Wq    = (const float*)d_in[8];
    const float* bq    = (const float*)d_in[9];
    const float* Wk    = (const float*)d_in[10];
    const float* bk    = (const float*)d_in[11];
    const float* Wv    = (const float*)d_in[12];
    const float* bv    = (const float*)d_in[13];

    float* out  = (float*)d_out;                         // (B,S,D)
    float* prob = out + (size_t)BATCH * SEQ * DIM;       // (B,H,S,S)

    const size_t perProj = (size_t)BATCH * HEADS * SEQ * DH;  // 2,097,152
    float* ws   = (float*)d_ws;
    float* qws  = ws;
    float* kws  = ws + perProj;
    float* vws  = ws + 2 * perProj;
    float* base = ws + 3 * perProj;                      // (B,S,S)

    // 1) projections -> (b,h,s,dh)
    proj_kernel<<<dim3(BATCH * SEQ / 16, DIM / 128, 3), 256, 0, stream>>>(
        query, key_t, value, Wq, Wk, Wv, bq, bk, bv, qws, kws, vws);

    // 2) head-independent rel/time blend term
    base_kernel<<<dim3(BATCH * SEQ), 256, 0, stream>>>(rel, tstmp, l1, l2, base);

    // 3) masked scores into prob region of d_out
    score_kernel<<<dim3(4, SEQ / 16, BATCH * HEADS), 256, 0, stream>>>(qws, kws, prob);

    // 4) softmax + blend (in place)
    softmax_combine_kernel<<<dim3(BATCH * HEADS * SEQ), 256, 0, stream>>>(prob, base, l1, l2);

    // 5) out = prob @ v
    out_kernel<<<dim3(4, DH / 16, BATCH * HEADS), 256, 0, stream>>>(prob, vws, out);
}